// ChunkedVectorizedSlidingWindowModel_4002909520749
// MI455X (gfx1250) — compile-verified
//
#include <hip/hip_runtime.h>
#include <math.h>

// ---------------------------------------------------------------------------
// Sliding-window attention (B=2, S=2048, D=512, H=8, Dh=64, W=128), fp32.
// All GEMM work routed through V_WMMA_F32_16X16X4_F32 (precision-preserving
// fp32 matrix path on CDNA5). Pipeline:
//   0) transpose the 4 weight matrices so every WMMA B-fragment is one
//      contiguous global_load_b64
//   1) fused QKV projection (WMMA GEMM; A-fragment reused across Wq/Wk/Wv;
//      K-loop chunked: prefetch once per 128-elt chunk, branchless fully-
//      unrolled software-pipelined inner loop); V written head-TRANSPOSED
//   2) flash-style windowed attention (WMMA scores + WMMA P*V, online
//      softmax; Q fragments register-resident across the 9 key tiles)
//   3) output projection (WMMA GEMM, same chunked/pipelined structure)
// Working set ~38 MB -> resident in the 192 MB L2; matrix-pipe bound.
// ---------------------------------------------------------------------------

typedef __attribute__((ext_vector_type(2))) float v2f;
typedef __attribute__((ext_vector_type(8))) float v8f;

#define D_MODEL 512
#define NHEAD   8
#define HEAD_DIM 64
#define WINDOW  128
#define SEQ     2048
#define BATCH   2
#define MROWS   (BATCH * SEQ)          // 4096
#define KCHUNK  128                    // prefetch granularity in K
#define NEG_BIG (-1.0e30f)             // finite "-inf" so m-rescale never NaNs

static __device__ __forceinline__ v8f wmma_f32(v2f a, v2f b, v8f c) {
  // 8 args: (neg_a, A, neg_b, B, c_mod, C, reuse_a, reuse_b)
  return __builtin_amdgcn_wmma_f32_16x16x4_f32(false, a, false, b, (short)0, c,
                                               false, false);
}

// ---------------------------------------------------------------------------
// Kernel 0: 512x512 transpose (LDS-tiled, conflict-free via +1 pad).
// ---------------------------------------------------------------------------
__global__ __launch_bounds__(256) void transpose512_kernel(
    const float* __restrict__ src, float* __restrict__ dst) {
  __shared__ float tile[32][33];
  const int tx = threadIdx.x & 31;
  const int ty = threadIdx.x >> 5;                 // 8 rows per pass
  const int bx = (blockIdx.x & 15) * 32;           // source col block
  const int by = (blockIdx.x >> 4) * 32;           // source row block
#pragma unroll
  for (int r = 0; r < 32; r += 8)
    tile[ty + r][tx] = src[(size_t)(by + ty + r) * D_MODEL + bx + tx];
  __syncthreads();
#pragma unroll
  for (int r = 0; r < 32; r += 8)
    dst[(size_t)(bx + ty + r) * D_MODEL + by + tx] = tile[tx][ty + r];
}

// ---------------------------------------------------------------------------
// Kernel 1: fused QKV projection with TRANSPOSED weights (Wt[n][k]).
// Q,K stored head-split [bh][s][d]; V stored head-transposed [bh][d][s].
// One wave per 16x16 output tile. Outer loop: 128-elt K chunks (one prefetch
// per chunk); inner loop: fully unrolled, branchless, double-buffered so the
// next step's 4 fragment loads are in flight during the current 3 WMMAs.
// ---------------------------------------------------------------------------
__global__ __launch_bounds__(256) void qkv_proj_kernel(
    const float* __restrict__ X,
    const float* __restrict__ Wtq, const float* __restrict__ Wtk,
    const float* __restrict__ Wtv,
    float* __restrict__ Qh, float* __restrict__ Kh, float* __restrict__ Vt) {
  const int lane = threadIdx.x & 31;
  const int wave = threadIdx.x >> 5;
  const int half = lane >> 4;   // 0: K=0,1 ; 1: K=2,3 (A/B fragment halves)
  const int l16  = lane & 15;

  const int rt = blockIdx.x >> 2;                    // row tile   0..255
  const int ct = ((blockIdx.x & 3) << 3) + wave;     // col tile   0..31

  const int arow = rt * 16 + l16;                    // A row for this lane
  const int col  = ct * 16 + l16;                    // output col for this lane

  v8f cq = {}; v8f ck = {}; v8f cv = {};

  const float* xrow = X + (size_t)arow * D_MODEL;
  const float* wq = Wtq + (size_t)col * D_MODEL;     // lane streams its own row
  const float* wk = Wtk + (size_t)col * D_MODEL;
  const float* wv = Wtv + (size_t)col * D_MODEL;

  const int kb = 2 * half;
  v2f a  = *(const v2f*)(xrow + kb);                 // prologue: fragments k=0
  v2f bq = *(const v2f*)(wq + kb);
  v2f bk = *(const v2f*)(wk + kb);
  v2f bv = *(const v2f*)(wv + kb);

#pragma unroll 1
  for (int kc = 0; kc < D_MODEL; kc += KCHUNK) {
    if (kc + KCHUNK < D_MODEL) {                     // once per chunk
      __builtin_prefetch(xrow + kc + KCHUNK, 0, 1);  // -> global_prefetch_b8
      __builtin_prefetch(wq + kc + KCHUNK, 0, 1);
      __builtin_prefetch(wk + kc + KCHUNK, 0, 1);
      __builtin_prefetch(wv + kc + KCHUNK, 0, 1);
    }
#pragma unroll
    for (int j = 0; j < KCHUNK; j += 4) {
      // Branchless next-step address: wraps to k=0 on the very last step
      // (harmless redundant load) so the inner body has no control flow.
      int kn = kc + j + 4;
      kn = (kn >= D_MODEL) ? 0 : kn;
      kn += kb;
      v2f an  = *(const v2f*)(xrow + kn);
      v2f bqn = *(const v2f*)(wq + kn);
      v2f bkn = *(const v2f*)(wk + kn);
      v2f bvn = *(const v2f*)(wv + kn);
      cq = wmma_f32(a, bq, cq);                      // consume current step
      ck = wmma_f32(a, bk, ck);
      cv = wmma_f32(a, bv, cv);
      a = an; bq = bqn; bk = bkn; bv = bvn;
    }
  }

  // C layout: VGPR i holds rows (i + 8*half), N = l16.
  const int h = col >> 6, d = col & 63;
#pragma unroll
  for (int i = 0; i < 8; ++i) {
    const int m = rt * 16 + i + 8 * half;            // global row (b*S + s)
    const int b = m >> 11, s = m & (SEQ - 1);
    const int bh = b * NHEAD + h;
    Qh[((size_t)bh * SEQ + s) * HEAD_DIM + d] = cq[i];
    Kh[((size_t)bh * SEQ + s) * HEAD_DIM + d] = ck[i];
    Vt[((size_t)bh * HEAD_DIM + d) * SEQ + s] = cv[i];   // head-transposed
  }
}

// ---------------------------------------------------------------------------
// Kernel 2: sliding-window attention, flash-style online softmax.
// One wave per (b, h, 16-query tile). 9 key tiles cover [q0-128, q0+15];
// tiles entirely below 0 are skipped (uniform branch -> EXEC stays all-ones).
// Q A-fragments are held in registers across all key tiles; P (16x16) is
// re-laid out C-fragment -> A-fragment via a per-wave LDS tile.
// ---------------------------------------------------------------------------
__global__ __launch_bounds__(128) void swa_kernel(
    const float* __restrict__ Qh, const float* __restrict__ Kh,
    const float* __restrict__ Vt, float* __restrict__ attn) {
  __shared__ float plds[4][16 * 16];   // 1 KB per wave

  const int lane = threadIdx.x & 31;
  const int wave = threadIdx.x >> 5;
  const int half = lane >> 4;
  const int l16  = lane & 15;
  float* P = plds[wave];

  const int task = blockIdx.x * 4 + wave;        // 0..2047
  const int qt   = task & 127;                   // S/16 tiles
  const int bh   = task >> 7;                    // 0..15  (b*H + h)
  const int q0   = qt * 16;

  const float* Qb  = Qh + (size_t)bh * SEQ * HEAD_DIM;
  const float* Kb  = Kh + (size_t)bh * SEQ * HEAD_DIM;
  const float* Vtb = Vt + (size_t)bh * HEAD_DIM * SEQ;   // [d][s]

  const int kb = 2 * half;

  // Q tile (16x64) register-resident: 16 A-fragments, reused for 9 key tiles.
  v2f qa[16];
#pragma unroll
  for (int kk = 0; kk < 16; ++kk)
    qa[kk] = *(const v2f*)(Qb + (size_t)(q0 + l16) * HEAD_DIM + kk * 4 + kb);

  v8f o0 = {}, o1 = {}, o2 = {}, o3 = {};        // O = 16x64 (4 C-fragments)
  float mrow[8], lrow[8];
#pragma unroll
  for (int i = 0; i < 8; ++i) { mrow[i] = NEG_BIG; lrow[i] = 0.0f; }

  const float scale = 0.125f;                    // 1/sqrt(64)

  for (int kt = q0 - WINDOW; kt <= q0; kt += 16) {
    if (kt + 15 < 0) continue;                   // wave-uniform skip (kt >= 0 below)

    const float* krow = Kb + (size_t)(kt + l16) * HEAD_DIM + kb;

    // ---- scores S = Q[16x64] * K^T[64x16], pipelined K loads -------------
    v8f s = {};
    v2f b = *(const v2f*)(krow);
#pragma unroll
    for (int kk = 0; kk < 16; ++kk) {            // fully static guards
      v2f bn = {};
      if (kk + 1 < 16) bn = *(const v2f*)(krow + (kk + 1) * 4);
      s = wmma_f32(qa[kk], b, s);
      b = bn;
    }

    // ---- mask + online softmax update ------------------------------------
#pragma unroll
    for (int i = 0; i < 8; ++i) {
      const int q   = q0 + i + 8 * half;         // this lane's query row
      const int key = kt + l16;                  // this lane's key column
      float sv = s[i] * scale;
      const bool valid = (key <= q) && (key >= q - (WINDOW - 1)) && (key >= 0);
      sv = valid ? sv : NEG_BIG;

      // row max across the 16 lanes of this half-wave
      float rm = sv;
      rm = fmaxf(rm, __shfl_xor(rm, 1, 16));
      rm = fmaxf(rm, __shfl_xor(rm, 2, 16));
      rm = fmaxf(rm, __shfl_xor(rm, 4, 16));
      rm = fmaxf(rm, __shfl_xor(rm, 8, 16));

      const float mnew  = fmaxf(mrow[i], rm);
      const float alpha = __expf(mrow[i] - mnew);   // finite: NEG_BIG sentinel
      const float p     = __expf(sv - mnew);        // masked -> exactly 0

      float rs = p;
      rs += __shfl_xor(rs, 1, 16);
      rs += __shfl_xor(rs, 2, 16);
      rs += __shfl_xor(rs, 4, 16);
      rs += __shfl_xor(rs, 8, 16);

      lrow[i] = lrow[i] * alpha + rs;
      mrow[i] = mnew;
      o0[i] *= alpha; o1[i] *= alpha; o2[i] *= alpha; o3[i] *= alpha;

      P[(i + 8 * half) * 16 + l16] = p;          // stash P tile (row-major)
    }
    // DS ops are in-order within a wave; backend inserts s_wait_dscnt for RAW.

    // ---- O += P[16x16] * V[16x64]  (Vt -> contiguous b64 B), pipelined ---
    {
      v2f a  = *(const v2f*)(P + l16 * 16 + kb);
      v2f b0 = *(const v2f*)(Vtb + (size_t)( 0 + l16) * SEQ + kt + kb);
      v2f b1 = *(const v2f*)(Vtb + (size_t)(16 + l16) * SEQ + kt + kb);
      v2f b2 = *(const v2f*)(Vtb + (size_t)(32 + l16) * SEQ + kt + kb);
      v2f b3 = *(const v2f*)(Vtb + (size_t)(48 + l16) * SEQ + kt + kb);
#pragma unroll
      for (int kk = 0; kk < 16; kk += 4) {       // fully static guards
        v2f an = {}, b0n = {}, b1n = {}, b2n = {}, b3n = {};
        if (kk + 4 < 16) {
          const int kn = kk + 4 + kb;
          an  = *(const v2f*)(P + l16 * 16 + kn);
          b0n = *(const v2f*)(Vtb + (size_t)( 0 + l16) * SEQ + kt + kn);
          b1n = *(const v2f*)(Vtb + (size_t)(16 + l16) * SEQ + kt + kn);
          b2n = *(const v2f*)(Vtb + (size_t)(32 + l16) * SEQ + kt + kn);
          b3n = *(const v2f*)(Vtb + (size_t)(48 + l16) * SEQ + kt + kn);
        }
        o0 = wmma_f32(a, b0, o0);
        o1 = wmma_f32(a, b1, o1);
        o2 = wmma_f32(a, b2, o2);
        o3 = wmma_f32(a, b3, o3);
        a = an; b0 = b0n; b1 = b1n; b2 = b2n; b3 = b3n;
      }
    }
  }

  // ---- normalize + write merged layout attn[b, s, h*64 + d] --------------
  const int b = bh >> 3, h = bh & 7;
#pragma unroll
  for (int i = 0; i < 8; ++i) {
    const float inv = 1.0f / lrow[i];
    const int srow = q0 + i + 8 * half;
    float* dst = attn + ((size_t)(b * SEQ + srow)) * D_MODEL + h * HEAD_DIM;
    dst[ 0 + l16] = o0[i] * inv;
    dst[16 + l16] = o1[i] * inv;
    dst[32 + l16] = o2[i] * inv;
    dst[48 + l16] = o3[i] * inv;
  }
}

// ---------------------------------------------------------------------------
// Kernel 3: output projection  Y = A[4096,512] * Wo^T-layout[512,512].
// Same chunked + branchless pipelined structure as the QKV GEMM.
// ---------------------------------------------------------------------------
__global__ __launch_bounds__(256) void out_proj_kernel(
    const float* __restrict__ A, const float* __restrict__ Wt,
    float* __restrict__ Y) {
  const int lane = threadIdx.x & 31;
  const int wave = threadIdx.x >> 5;
  const int half = lane >> 4;
  const int l16  = lane & 15;

  const int rt = blockIdx.x >> 2;
  const int ct = ((blockIdx.x & 3) << 3) + wave;

  const int arow = rt * 16 + l16;
  const int col  = ct * 16 + l16;

  v8f c = {};
  const float* ar = A + (size_t)arow * D_MODEL;
  const float* wr = Wt + (size_t)col * D_MODEL;

  const int kb = 2 * half;
  v2f a = *(const v2f*)(ar + kb);
  v2f b = *(const v2f*)(wr + kb);

#pragma unroll 1
  for (int kc = 0; kc < D_MODEL; kc += KCHUNK) {
    if (kc + KCHUNK < D_MODEL) {
      __builtin_prefetch(ar + kc + KCHUNK, 0, 1);
      __builtin_prefetch(wr + kc + KCHUNK, 0, 1);
    }
#pragma unroll
    for (int j = 0; j < KCHUNK; j += 4) {
      int kn = kc + j + 4;
      kn = (kn >= D_MODEL) ? 0 : kn;               // branchless wrap
      kn += kb;
      v2f an = *(const v2f*)(ar + kn);
      v2f bn = *(const v2f*)(wr + kn);
      c = wmma_f32(a, b, c);
      a = an; b = bn;
    }
  }

#pragma unroll
  for (int i = 0; i < 8; ++i) {
    const int m = rt * 16 + i + 8 * half;
    Y[(size_t)m * D_MODEL + col] = c[i];
  }
}

// ---------------------------------------------------------------------------
extern "C" void kernel_launch(void* const* d_in, const int* in_sizes, int n_in,
                              void* d_out, int out_size, void* d_ws,
                              size_t ws_size, hipStream_t stream) {
  const float* x  = (const float*)d_in[0];
  const float* Wq = (const float*)d_in[1];
  const float* Wk = (const float*)d_in[2];
  const float* Wv = (const float*)d_in[3];
  const float* Wo = (const float*)d_in[4];

  // Workspace (fp32): Qh | Kh | Vt | attn (8 MB each) | 4 transposed weights.
  float* Qh   = (float*)d_ws;
  float* Kh   = Qh + (size_t)MROWS * D_MODEL;
  float* Vt   = Kh + (size_t)MROWS * D_MODEL;
  float* attn = Vt + (size_t)MROWS * D_MODEL;
  float* Wtq  = attn + (size_t)MROWS * D_MODEL;
  float* Wtk  = Wtq + (size_t)D_MODEL * D_MODEL;
  float* Wtv  = Wtk + (size_t)D_MODEL * D_MODEL;
  float* Wto  = Wtv + (size_t)D_MODEL * D_MODEL;

  // Stage 0: weight transposes (256 blocks of 32x32 tiles each).
  transpose512_kernel<<<256, 256, 0, stream>>>(Wq, Wtq);
  transpose512_kernel<<<256, 256, 0, stream>>>(Wk, Wtk);
  transpose512_kernel<<<256, 256, 0, stream>>>(Wv, Wtv);
  transpose512_kernel<<<256, 256, 0, stream>>>(Wo, Wto);

  // 256 row-tiles * 32 col-tiles = 8192 wave-tiles; 8 waves/block.
  qkv_proj_kernel<<<1024, 256, 0, stream>>>(x, Wtq, Wtk, Wtv, Qh, Kh, Vt);

  // 2 * 8 * 128 = 2048 wave-tasks; 4 waves/block.
  swa_kernel<<<512, 128, 0, stream>>>(Qh, Kh, Vt, attn);

  out_proj_kernel<<<1024, 256, 0, stream>>>(attn, Wto, (float*)d_out);
}